// HPR_D_72353019068523
// MI455X (gfx1250) — compile-verified
//
#include <hip/hip_runtime.h>
#include <hip/hip_bf16.h>
#include <math.h>

typedef float v2f __attribute__((ext_vector_type(2)));
typedef float v8f __attribute__((ext_vector_type(8)));

#define DDIM 256
#define KCLS 5
#define PITCH 260          // 260 % 64 = 4 -> rows hit distinct LDS banks
#define MAXN  9.96f        // (1 - EPS_PROJ)/SQRT_C

__device__ __forceinline__ float waveRedSum(float v) {
#pragma unroll
  for (int off = 16; off > 0; off >>= 1) v += __shfl_xor(v, off, 32);
  return v;
}

// ---------------- Kernel 1: support rows -> proto_p, num/den init ----------------
__global__ __launch_bounds__(256) void hpr_k1(const float* __restrict__ feat,
                                              float* __restrict__ proto,
                                              float* __restrict__ proto_y2,
                                              float* __restrict__ num,
                                              float* __restrict__ den,
                                              float* __restrict__ lossA) {
  __shared__ float red[256];
  int t = threadIdx.x;
  float agk[KCLS], ag[KCLS];
#pragma unroll
  for (int c = 0; c < KCLS; c++) { agk[c] = 0.f; ag[c] = 0.f; }

  for (int r = 0; r < 25; r++) {
    float v = feat[r * DDIM + t];
    red[t] = v * v; __syncthreads();
    for (int s = 128; s > 0; s >>= 1) { if (t < s) red[t] += red[t + s]; __syncthreads(); }
    float n2 = red[0]; __syncthreads();
    float n  = fmaxf(sqrtf(n2), 1e-5f);
    float sc = tanhf(0.1f * n) / (0.1f * n);       // expmap0 scale
    float pn = sc * n;
    if (pn > MAXN) { sc *= MAXN / pn; pn = MAXN; } // project
    float x2    = pn * pn;
    float kq    = 2.0f / (1.0f + 0.01f * x2);      // p2k scale
    float gamma = rsqrtf(1.0f - 0.01f * kq * kq * x2);
    int c = r / 5;
    agk[c] += gamma * kq * sc * v;                 // gamma_s * klein_s element
    ag[c]  += gamma;
  }

  for (int c = 0; c < KCLS; c++) {
    num[c * DDIM + t] = agk[c];                    // init num with support sums
    float mk = agk[c] / ag[c];                     // Klein mean
    red[t] = mk * mk; __syncthreads();
    for (int s = 128; s > 0; s >>= 1) { if (t < s) red[t] += red[t + s]; __syncthreads(); }
    float mk2 = red[0]; __syncthreads();
    float dn = 1.0f + sqrtf(1.0f - 0.01f * mk2);   // k2p
    proto[c * DDIM + t] = mk / dn;
    if (t == 0) { proto_y2[c] = mk2 / (dn * dn); den[c] = ag[c]; }
  }
  if (t == 0) lossA[0] = 0.f;
}

// ---------------- Main pass kernel: 2 waves, 16 queries per wave ----------------
template <bool PASS2>
__global__ __launch_bounds__(64) void hpr_pass(const float* __restrict__ feat,
                                               const int* __restrict__ label,
                                               const float* __restrict__ proto,
                                               const float* __restrict__ proto_y2,
                                               float* __restrict__ num,
                                               float* __restrict__ den,
                                               float* __restrict__ lossA,
                                               float* __restrict__ out, int Q) {
  __shared__ __align__(16) float tile[2][16][PITCH];   // queries_p tiles
  __shared__ __align__(16) float sproto[16][PITCH];    // protos zero-padded to 16 rows
  __shared__ float sX2[2][16];
  __shared__ float sDist[2][16][8];
  __shared__ int   sPred[32];
  __shared__ float sCoef[32];
  __shared__ float bden[KCLS];
  __shared__ float bloss;

  const int tid  = threadIdx.x;
  const int wave = tid >> 5, lane = tid & 31;
  const int m16  = lane & 15, half = lane >> 4;
  const int qbase = blockIdx.x * 32 + wave * 16;

  // stage zero-padded protos into LDS (B matrix source)
  for (int idx = tid; idx < 16 * DDIM; idx += 64) {
    int r = idx >> 8, cc = idx & 255;
    sproto[r][cc] = (r < KCLS) ? proto[r * DDIM + cc] : 0.f;
  }
  if (!PASS2) {
    if (tid < KCLS) bden[tid] = 0.f;
    if (tid == KCLS) bloss = 0.f;
  }

  // coalesced load of 16 query rows (b128 per lane)
  for (int r = 0; r < 16; r++) {
    int q = qbase + r;
    const float* row = feat + (size_t)(25 + (q < Q ? q : 0)) * DDIM;
    float4 a0 = ((const float4*)row)[lane];
    float4 a1 = ((const float4*)row)[lane + 32];
    *(float4*)&tile[wave][r][lane * 4]       = a0;
    *(float4*)&tile[wave][r][128 + lane * 4] = a1;
  }
  __syncthreads();

  // per-row norm -> scale tile in place to queries_p
  for (int r = 0; r < 16; r++) {
    float acc = 0.f;
#pragma unroll
    for (int i = 0; i < 8; i++) { float v = tile[wave][r][i * 32 + lane]; acc += v * v; }
    float n2 = waveRedSum(acc);
    float n  = fmaxf(sqrtf(n2), 1e-5f);
    float sc = tanhf(0.1f * n) / (0.1f * n);
    float pn = sc * n;
    if (pn > MAXN) { sc *= MAXN / pn; pn = MAXN; }
#pragma unroll
    for (int i = 0; i < 8; i++) tile[wave][r][i * 32 + lane] *= sc;
    if (lane == 0) sX2[wave][r] = pn * pn;
  }
  __syncthreads();

  // ---- WMMA: xy = queries_p(16xD) . protos^T(Dx16), K-chunks of 4 ----
  v8f c = {0.f, 0.f, 0.f, 0.f, 0.f, 0.f, 0.f, 0.f};
  for (int kc = 0; kc < DDIM; kc += 4) {
    int col = kc + half * 2;
    v2f a = *(const v2f*)&tile[wave][m16][col];    // A: 16x4 f32 fragment
    v2f b = *(const v2f*)&sproto[m16][col];        // B: 4x16 f32 fragment (N=m16)
    c = __builtin_amdgcn_wmma_f32_16x16x4_f32(false, a, false, b, (short)0, c,
                                              false, false);
  }

  // closed-form hyperbolic distance from (x2, y2, xy)
  float y2 = proto_y2[(m16 < KCLS) ? m16 : 0];
  if (m16 < KCLS) {
#pragma unroll
    for (int g = 0; g < 8; g++) {
      int   m  = 8 * half + g;
      float x2 = sX2[wave][m];
      float xy = c[g];
      float A  = 1.f - 0.02f * xy + 0.01f * y2;
      float B  = 1.f - 0.01f * x2;
      float nv2 = fmaxf(A * A * x2 - 2.f * A * B * xy + B * B * y2, 0.f);
      float dnm = 1.f - 0.02f * xy + 1e-4f * x2 * y2 + 1e-5f;
      float man = sqrtf(nv2) / dnm;
      float z = fminf(fmaxf(0.1f * man, -1.f + 1e-5f), 1.f - 1e-5f);
      sDist[wave][m][m16] = 10.f * (log1pf(z) - log1pf(-z)); // 2/sqrt(C)*artanh
    }
  }
  __syncthreads();

  // ---- per-query finalize (lanes 0..15, query m = lane) ----
  float ceContrib = 0.f;
  if (lane < 16) {
    int  q = qbase + lane;
    bool valid = (q < Q);
    float dv[KCLS];
#pragma unroll
    for (int k = 0; k < KCLS; k++) dv[k] = sDist[wave][lane][k];
    float best = dv[0]; int pred = 0;
#pragma unroll
    for (int k = 1; k < KCLS; k++) { if (dv[k] < best) { best = dv[k]; pred = k; } }

    if (PASS2) {
      float e[KCLS]; float se = 0.f;
#pragma unroll
      for (int k = 0; k < KCLS; k++) { e[k] = expf(best - dv[k]); se += e[k]; }
      if (valid) {
        float inv = 1.f / se;
#pragma unroll
        for (int k = 0; k < KCLS; k++) out[(size_t)q * KCLS + k] = e[k] * inv;
      }
    } else {
      float se = 0.f;
#pragma unroll
      for (int k = 0; k < KCLS; k++) se += expf(best - dv[k]);
      float lse = logf(se);
      int lbl = label[Q + q];                       // label[1][q]
      float dl = dv[0];
#pragma unroll
      for (int k = 1; k < KCLS; k++) dl = (lbl == k) ? dv[k] : dl;
      float ce = (best - dl) - lse;                 // log_softmax is idempotent
      ceContrib = valid ? ce : 0.f;

      float x2q   = sX2[wave][lane];
      float kq    = 2.0f / (1.0f + 0.01f * x2q);
      float gamma = rsqrtf(1.0f - 0.01f * kq * kq * x2q);
      sPred[wave * 16 + lane] = pred;
      sCoef[wave * 16 + lane] = valid ? gamma * kq : 0.f;
      if (valid) atomicAdd(&bden[pred], gamma);
    }
  }

  if (!PASS2) {
    float s = waveRedSum(ceContrib);
    if (lane == 0) atomicAdd(&bloss, s);
    __syncthreads();

    // class-masked accumulation of gamma*klein into num (each thread: 4 dims)
    float racc[KCLS][4];
#pragma unroll
    for (int cc = 0; cc < KCLS; cc++)
#pragma unroll
      for (int j = 0; j < 4; j++) racc[cc][j] = 0.f;

    for (int m2 = 0; m2 < 32; m2++) {
      int p = sPred[m2]; float cf = sCoef[m2];
      int wv = m2 >> 4, r = m2 & 15;
      float v0 = tile[wv][r][tid];
      float v1 = tile[wv][r][tid + 64];
      float v2 = tile[wv][r][tid + 128];
      float v3 = tile[wv][r][tid + 192];
#pragma unroll
      for (int cc = 0; cc < KCLS; cc++) {
        float msk = (p == cc) ? cf : 0.f;
        racc[cc][0] += msk * v0; racc[cc][1] += msk * v1;
        racc[cc][2] += msk * v2; racc[cc][3] += msk * v3;
      }
    }
#pragma unroll
    for (int cc = 0; cc < KCLS; cc++) {
      atomicAdd(&num[cc * DDIM + tid],       racc[cc][0]);
      atomicAdd(&num[cc * DDIM + tid + 64],  racc[cc][1]);
      atomicAdd(&num[cc * DDIM + tid + 128], racc[cc][2]);
      atomicAdd(&num[cc * DDIM + tid + 192], racc[cc][3]);
    }
    if (tid < KCLS) atomicAdd(&den[tid], bden[tid]);
    if (tid == KCLS) atomicAdd(lossA, bloss);
  }
}

// ---------------- Kernel 3: proto_new = k2p(num/den), finalize loss ----------------
__global__ __launch_bounds__(256) void hpr_k3(const float* __restrict__ num,
                                              const float* __restrict__ den,
                                              const float* __restrict__ lossA,
                                              float* __restrict__ proto2,
                                              float* __restrict__ proto2_y2,
                                              float* __restrict__ out, int Q) {
  __shared__ float red[256];
  int t = threadIdx.x;
  for (int c = 0; c < KCLS; c++) {
    float mk = num[c * DDIM + t] / den[c];
    red[t] = mk * mk; __syncthreads();
    for (int s = 128; s > 0; s >>= 1) { if (t < s) red[t] += red[t + s]; __syncthreads(); }
    float mk2 = red[0]; __syncthreads();
    float dn = 1.0f + sqrtf(1.0f - 0.01f * mk2);
    proto2[c * DDIM + t] = mk / dn;
    if (t == 0) proto2_y2[c] = mk2 / (dn * dn);
  }
  if (t == 0) out[(size_t)Q * KCLS] = -lossA[0] / (float)Q;
}

extern "C" void kernel_launch(void* const* d_in, const int* in_sizes, int n_in,
                              void* d_out, int out_size, void* d_ws, size_t ws_size,
                              hipStream_t stream) {
  const float* feat  = (const float*)d_in[0];
  const int*   label = (const int*)d_in[1];
  float* out = (float*)d_out;
  float* ws  = (float*)d_ws;

  int total_rows = in_sizes[0] / DDIM;
  int Q = total_rows - 25;

  float* proto     = ws;            // 1280
  float* proto_y2  = ws + 1280;     // 5 (pad to 8)
  float* num       = ws + 1288;     // 1280
  float* den       = ws + 2568;     // 5 (pad)
  float* lossA     = ws + 2576;     // 1 (pad)
  float* proto2    = ws + 2580;     // 1280
  float* proto2_y2 = ws + 3860;     // 5

  hpr_k1<<<1, 256, 0, stream>>>(feat, proto, proto_y2, num, den, lossA);

  int nblk = (Q + 31) / 32;
  hpr_pass<false><<<nblk, 64, 0, stream>>>(feat, label, proto, proto_y2,
                                           num, den, lossA, out, Q);
  hpr_k3<<<1, 256, 0, stream>>>(num, den, lossA, proto2, proto2_y2, out, Q);
  hpr_pass<true><<<nblk, 64, 0, stream>>>(feat, label, proto2, proto2_y2,
                                          num, den, lossA, out, Q);
}